// Model_9380208574653
// MI455X (gfx1250) — compile-verified
//
#include <hip/hip_runtime.h>
#include <hip/hip_bf16.h>
#include <math.h>

typedef __attribute__((ext_vector_type(2))) float v2f;
typedef __attribute__((ext_vector_type(8))) float v8f;

#define B_NUM 1024
#define E_NUM 50000
#define R_NUM 500
#define D_DIM 512
#define P_DIM 16
#define C_NUM 8
#define E_TILES (E_NUM / 16)   // 3125

// ---------------------------------------------------------------------------
// init: zero the two factor accumulators (d_ws is not re-poisoned per call)
// ---------------------------------------------------------------------------
__global__ void init_kernel(float* accs) {
    if (threadIdx.x < 2) accs[threadIdx.x] = 0.0f;
}

// ---------------------------------------------------------------------------
// gather hr = concat(entity_w[head], relation_w[rel])  -> [B, 1024]
// ---------------------------------------------------------------------------
__global__ __launch_bounds__(256) void gather_hr(const int* __restrict__ heads,
                                                 const int* __restrict__ rels,
                                                 const float* __restrict__ entity_w,
                                                 const float* __restrict__ relation_w,
                                                 float* __restrict__ hr) {
    const int b = blockIdx.x;
    const int tid = threadIdx.x;
    const int h = heads[b];
    const int r = rels[b];
    const float4* hp = (const float4*)(entity_w + (size_t)h * D_DIM);
    const float4* rp = (const float4*)(relation_w + (size_t)r * D_DIM);
    float4* out = (float4*)(hr + (size_t)b * (2 * D_DIM));
    if (tid < 128) out[tid] = hp[tid];
    else           out[tid] = rp[tid - 128];
}

// ---------------------------------------------------------------------------
// hid = relu(hr @ attn_w1 + b1)  :  [1024,1024] @ [1024,512]
// fp32 WMMA 16x16x4. One block per 16-row batch tile; A staged in LDS in two
// 512-column passes (32KB); each of 8 waves owns 4 n-tiles (64 columns).
// ---------------------------------------------------------------------------
__global__ __launch_bounds__(256) void hid_gemm(const float* __restrict__ hr,
                                                const float* __restrict__ w1,
                                                const float* __restrict__ b1,
                                                float* __restrict__ hid) {
    __shared__ float shA[16 * D_DIM];           // 32 KB
    const int bm   = blockIdx.x;                // 0..63
    const int tid  = threadIdx.x;
    const int wave = tid >> 5;
    const int lane = tid & 31;
    const int m    = lane & 15;
    const int g    = lane >> 4;                 // 0 or 1
    const int nBase0 = wave * 64;

    v8f c[4] = {};
    for (int pass = 0; pass < 2; ++pass) {
        __syncthreads();                        // protect LDS reuse across passes
        for (int i = tid; i < 16 * D_DIM / 4; i += 256) {
            const int row  = i >> 7;            // 128 float4 per row
            const int col4 = i & 127;
            ((float4*)shA)[i] =
                ((const float4*)(hr + (size_t)(bm * 16 + row) * (2 * D_DIM)
                                    + pass * D_DIM))[col4];
        }
        __syncthreads();
        const float* Ap = shA + m * D_DIM + 2 * g;
        for (int k = 0; k < D_DIM; k += 4) {
            const v2f a = *(const v2f*)(Ap + k);
            const int kg = pass * D_DIM + k + 2 * g;   // global K of VGPR0
#pragma unroll
            for (int nt = 0; nt < 4; ++nt) {
                const int n = nBase0 + nt * 16 + m;
                v2f bfrag = { w1[(size_t)kg * D_DIM + n],
                              w1[(size_t)(kg + 1) * D_DIM + n] };
                c[nt] = __builtin_amdgcn_wmma_f32_16x16x4_f32(
                    false, a, false, bfrag, (short)0, c[nt], false, false);
            }
        }
    }
#pragma unroll
    for (int nt = 0; nt < 4; ++nt) {
        const int n = nBase0 + nt * 16 + m;
        const float bias = b1[n];
#pragma unroll
        for (int j = 0; j < 8; ++j) {
            const int row = bm * 16 + j + 8 * g;
            float v = c[nt][j] + bias;
            hid[(size_t)row * D_DIM + n] = v > 0.0f ? v : 0.0f;
        }
    }
}

// ---------------------------------------------------------------------------
// attn = softmax(hid @ attn_w2 + b2)   [1024,512] @ [512,8]
// one thread per batch row; w2 staged in LDS
// ---------------------------------------------------------------------------
__global__ __launch_bounds__(256) void attn_kernel(const float* __restrict__ hid,
                                                   const float* __restrict__ w2,
                                                   const float* __restrict__ b2,
                                                   float* __restrict__ attn_ws,
                                                   float* __restrict__ attn_out) {
    __shared__ float shW[D_DIM * C_NUM];        // 16 KB
    const int tid = threadIdx.x;
    for (int i = tid; i < D_DIM * C_NUM; i += 256) shW[i] = w2[i];
    __syncthreads();

    const int b = blockIdx.x * 256 + tid;       // grid = 4 blocks
    float acc[C_NUM];
#pragma unroll
    for (int c = 0; c < C_NUM; ++c) acc[c] = b2[c];
    const float* hp = hid + (size_t)b * D_DIM;
    for (int j = 0; j < D_DIM; ++j) {
        const float h = hp[j];
#pragma unroll
        for (int c = 0; c < C_NUM; ++c) acc[c] += h * shW[j * C_NUM + c];
    }
    float mx = acc[0];
#pragma unroll
    for (int c = 1; c < C_NUM; ++c) mx = fmaxf(mx, acc[c]);
    float s = 0.0f;
#pragma unroll
    for (int c = 0; c < C_NUM; ++c) { acc[c] = expf(acc[c] - mx); s += acc[c]; }
    const float inv = 1.0f / s;
#pragma unroll
    for (int c = 0; c < C_NUM; ++c) {
        const float v = acc[c] * inv;
        attn_ws[(size_t)b * C_NUM + c]  = v;
        attn_out[(size_t)b * C_NUM + c] = v;
    }
}

// ---------------------------------------------------------------------------
// per-triple core contraction + norm accumulation (Q = 1 -> plain sumsq)
// one block per batch element
// ---------------------------------------------------------------------------
__global__ __launch_bounds__(256) void triple_kernel(const int* __restrict__ heads,
                                                     const int* __restrict__ rels,
                                                     const int* __restrict__ tails,
                                                     const float* __restrict__ entity_w,
                                                     const float* __restrict__ relation_w,
                                                     const float* __restrict__ W_cores,
                                                     const float* __restrict__ attn_ws,
                                                     float* __restrict__ x1_out,
                                                     float* __restrict__ accs) {
    __shared__ float shW[4096];                 // Weff [P,P,P], 16 KB
    __shared__ float shH[D_DIM], shR[D_DIM], shT[D_DIM];
    __shared__ float red[2];
    const int b   = blockIdx.x;
    const int tid = threadIdx.x;
    const int hI = heads[b], rI = rels[b], tI = tails[b];

    float f1 = 0.0f;
    for (int i = tid; i < D_DIM; i += 256) {
        const float hv = entity_w[(size_t)hI * D_DIM + i];
        const float rv = relation_w[(size_t)rI * D_DIM + i];
        const float tv = entity_w[(size_t)tI * D_DIM + i];
        shH[i] = hv; shR[i] = rv; shT[i] = tv;
        f1 += hv * hv + rv * rv + tv * tv;
    }
    float a[C_NUM];
#pragma unroll
    for (int c = 0; c < C_NUM; ++c) a[c] = attn_ws[(size_t)b * C_NUM + c];
    for (int i = tid; i < 4096; i += 256) {
        float w = 0.0f;
#pragma unroll
        for (int c = 0; c < C_NUM; ++c) w += a[c] * W_cores[c * 4096 + i];
        shW[i] = w;
    }
    if (tid == 0) { red[0] = 0.0f; red[1] = 0.0f; }
    __syncthreads();

    float f2 = 0.0f;
    // x1[b,i,s] = sum_{p,q} h[i,p] r[i,q] W[p,q,s]  (stored for scores GEMM)
    for (int pair = tid; pair < 512; pair += 256) {
        const int i = pair >> 4, s = pair & 15;
        const float* h = shH + i * 16;
        const float* r = shR + i * 16;
        float acc = 0.0f;
        for (int p = 0; p < 16; ++p) {
            const float hp = h[p];
            const float* Wp = shW + p * 256 + s;
#pragma unroll
            for (int q = 0; q < 16; ++q) acc += hp * r[q] * Wp[q * 16];
        }
        x1_out[(size_t)b * D_DIM + pair] = acc;
        f2 += acc * acc;
    }
    // x2[b,i,p] = sum_{q,s} r[i,q] t[i,s] W[p,q,s]
    for (int pair = tid; pair < 512; pair += 256) {
        const int i = pair >> 4, p = pair & 15;
        const float* r = shR + i * 16;
        const float* t = shT + i * 16;
        float acc = 0.0f;
        for (int q = 0; q < 16; ++q) {
            const float rq = r[q];
            const float* Wq = shW + p * 256 + q * 16;
#pragma unroll
            for (int s = 0; s < 16; ++s) acc += rq * t[s] * Wq[s];
        }
        f2 += acc * acc;
    }
    // x3[b,i,q] = sum_{p,s} h[i,p] t[i,s] W[p,q,s]
    for (int pair = tid; pair < 512; pair += 256) {
        const int i = pair >> 4, q = pair & 15;
        const float* h = shH + i * 16;
        const float* t = shT + i * 16;
        float acc = 0.0f;
        for (int p = 0; p < 16; ++p) {
            const float hp = h[p];
            const float* Wp = shW + p * 256 + q * 16;
#pragma unroll
            for (int s = 0; s < 16; ++s) acc += hp * t[s] * Wp[s];
        }
        f2 += acc * acc;
    }
    atomicAdd(&red[0], f1);
    atomicAdd(&red[1], f2);
    __syncthreads();
    if (tid == 0) {
        atomicAdd(&accs[0], red[0]);
        atomicAdd(&accs[1], red[1]);
    }
}

// ---------------------------------------------------------------------------
// geometric consistency over the 8 cores + finalize factor1/factor2
// ---------------------------------------------------------------------------
__global__ __launch_bounds__(256) void geom_kernel(const float* __restrict__ W_cores,
                                                   const float* __restrict__ accs,
                                                   float* __restrict__ factors) {
    __shared__ float sums[C_NUM][4];
    __shared__ float centers[C_NUM][3];
    const int tid = threadIdx.x;
    const int c = tid >> 5, lane = tid & 31;
    if (lane < 4) sums[c][lane] = 0.0f;
    __syncthreads();
    float tot = 0.0f, sx = 0.0f, sy = 0.0f, sz = 0.0f;
    for (int idx = lane; idx < 4096; idx += 32) {
        const float w = fabsf(W_cores[c * 4096 + idx]);
        const int p = idx >> 8, q = (idx >> 4) & 15, s = idx & 15;
        tot += w; sx += w * (float)p; sy += w * (float)q; sz += w * (float)s;
    }
    atomicAdd(&sums[c][0], tot);
    atomicAdd(&sums[c][1], sx);
    atomicAdd(&sums[c][2], sy);
    atomicAdd(&sums[c][3], sz);
    __syncthreads();
    if (tid < C_NUM) {
        const float t = sums[tid][0];
        centers[tid][0] = sums[tid][1] / t;
        centers[tid][1] = sums[tid][2] / t;
        centers[tid][2] = sums[tid][3] / t;
    }
    __syncthreads();
    if (tid == 0) {
        float f3 = 0.0f;
        for (int i = 0; i < C_NUM; ++i) {
            float mind = 1e30f;
            for (int j = 0; j < C_NUM; ++j) {
                if (j == i) continue;
                const float dx = centers[i][0] - centers[j][0];
                const float dy = centers[i][1] - centers[j][1];
                const float dz = centers[i][2] - centers[j][2];
                mind = fminf(mind, sqrtf(dx * dx + dy * dy + dz * dz));
            }
            f3 += logf(mind + 1e-6f);
        }
        factors[0] = accs[0] / (float)B_NUM;    // factor1
        factors[1] = accs[1] / (float)B_NUM;    // factor2
        factors[2] = -f3 / (float)C_NUM;        // factor3
    }
}

// ---------------------------------------------------------------------------
// scores = x1 @ entity_w.T : [1024,512] @ [512,50000]  (dominant GEMM)
// fp32 WMMA 16x16x4. grid.y = 64 batch tiles; each block covers 32 e-tiles
// (8 waves x 4 tiles per wave sharing one A fragment per k-step:
//  1 ds_load_b64 + 4 global_load_b64 -> 4 wmma). A tile staged in LDS.
// Out-of-range e-tiles clamp their row (loads stay legal, EXEC stays uniform)
// and skip the store.
// ---------------------------------------------------------------------------
__global__ __launch_bounds__(256) void scores_gemm(const float* __restrict__ x1,
                                                   const float* __restrict__ entity_w,
                                                   float* __restrict__ scores) {
    __shared__ float shA[16 * D_DIM];           // 32 KB
    const int bm  = blockIdx.y;                 // 0..63
    const int tid = threadIdx.x;

    const float4* src = (const float4*)(x1 + (size_t)bm * 16 * D_DIM);
    for (int i = tid; i < 16 * D_DIM / 4; i += 256) ((float4*)shA)[i] = src[i];
    __syncthreads();

    const int wave = tid >> 5;
    const int lane = tid & 31;
    const int m = lane & 15;
    const int g = lane >> 4;
    const int eTileBase = blockIdx.x * 32 + wave * 4;   // wave-uniform

    const float* Bp[4];
    bool valid[4];
#pragma unroll
    for (int nt = 0; nt < 4; ++nt) {
        int et = eTileBase + nt;
        valid[nt] = (et < E_TILES);
        if (et >= E_TILES) et = E_TILES - 1;            // clamp: legal address
        Bp[nt] = entity_w + (size_t)(et * 16 + m) * D_DIM + 2 * g;
    }
    const float* Ap = shA + m * D_DIM + 2 * g;

    v8f c[4] = {};
    for (int k = 0; k < D_DIM; k += 4) {
        const v2f a = *(const v2f*)(Ap + k);            // ds_load_b64 (shared)
#pragma unroll
        for (int nt = 0; nt < 4; ++nt) {
            const v2f b = *(const v2f*)(Bp[nt] + k);    // global_load_b64
            c[nt] = __builtin_amdgcn_wmma_f32_16x16x4_f32(
                false, a, false, b, (short)0, c[nt], false, false);
        }
    }
#pragma unroll
    for (int nt = 0; nt < 4; ++nt) {
        if (!valid[nt]) continue;
        const size_t eCol = (size_t)(eTileBase + nt) * 16 + m;
#pragma unroll
        for (int j = 0; j < 8; ++j) {
            const int row = bm * 16 + j + 8 * g;
            scores[(size_t)row * E_NUM + eCol] = c[nt][j];
        }
    }
}

// ---------------------------------------------------------------------------
extern "C" void kernel_launch(void* const* d_in, const int* in_sizes, int n_in,
                              void* d_out, int out_size, void* d_ws, size_t ws_size,
                              hipStream_t stream) {
    const int*   heads      = (const int*)d_in[0];
    const int*   rels       = (const int*)d_in[1];
    const int*   tails      = (const int*)d_in[2];
    const float* entity_w   = (const float*)d_in[3];
    const float* relation_w = (const float*)d_in[4];
    const float* W_cores    = (const float*)d_in[5];
    const float* attn_w1    = (const float*)d_in[6];
    const float* attn_b1    = (const float*)d_in[7];
    const float* attn_w2    = (const float*)d_in[8];
    const float* attn_b2    = (const float*)d_in[9];

    float* ws      = (float*)d_ws;
    float* hr_ws   = ws;                                   // B * 2D
    float* hid_ws  = hr_ws  + (size_t)B_NUM * 2 * D_DIM;   // B * D
    float* x1_ws   = hid_ws + (size_t)B_NUM * D_DIM;       // B * D
    float* attn_ws = x1_ws  + (size_t)B_NUM * D_DIM;       // B * C
    float* accs    = attn_ws + (size_t)B_NUM * C_NUM;      // 2

    float* scores   = (float*)d_out;
    float* factors  = scores + (size_t)B_NUM * E_NUM;      // 3 scalars
    float* attn_out = factors + 3;                         // B * C

    init_kernel<<<1, 64, 0, stream>>>(accs);
    gather_hr<<<B_NUM, 256, 0, stream>>>(heads, rels, entity_w, relation_w, hr_ws);
    hid_gemm<<<B_NUM / 16, 256, 0, stream>>>(hr_ws, attn_w1, attn_b1, hid_ws);
    attn_kernel<<<B_NUM / 256, 256, 0, stream>>>(hid_ws, attn_w2, attn_b2,
                                                 attn_ws, attn_out);
    triple_kernel<<<B_NUM, 256, 0, stream>>>(heads, rels, tails, entity_w,
                                             relation_w, W_cores, attn_ws,
                                             x1_ws, accs);
    geom_kernel<<<1, 256, 0, stream>>>(W_cores, accs, factors);

    dim3 g((E_TILES + 31) / 32, B_NUM / 16);
    scores_gemm<<<g, 256, 0, stream>>>(x1_ws, entity_w, scores);
}